// DimeNetStyleAngleFeatureExtractor_15607911153874
// MI455X (gfx1250) — compile-verified
//
#include <hip/hip_runtime.h>
#include <math.h>

#define CUTOFF_F 5.0f
#define NR 6
#define NS 7
#define KDIM 42        // NS * NR
#define KPAD 44        // padded to multiple of 4 for WMMA K-steps
#define H 128
#define EPSF 1e-6f
#define PI_F 3.14159265358979323846f

typedef __attribute__((ext_vector_type(2))) float v2f;
typedef __attribute__((ext_vector_type(8))) float v8f;

// ---------------------------------------------------------------------------
// Kernel 1: per-edge Bessel RBF -> ws  (rbf[e][0..5])
// ---------------------------------------------------------------------------
__global__ void rbf_kernel(const float* __restrict__ node_pos,
                           const int* __restrict__ ei,
                           const int* __restrict__ ej,
                           float* __restrict__ rbf, int E) {
    int e = blockIdx.x * blockDim.x + threadIdx.x;
    if (e >= E) return;
    int i = ei[e], j = ej[e];
    float dx = node_pos[3 * i + 0] - node_pos[3 * j + 0];
    float dy = node_pos[3 * i + 1] - node_pos[3 * j + 1];
    float dz = node_pos[3 * i + 2] - node_pos[3 * j + 2];
    float d = sqrtf(dx * dx + dy * dy + dz * dz);
    d = fmaxf(d, EPSF);
    float ds = d / CUTOFF_F;
    float env = (ds <= 1.0f) ? 0.5f * (cosf(PI_F * ds) + 1.0f) : 0.0f;
    float scale = env / fmaxf(ds, EPSF);
#pragma unroll
    for (int n = 0; n < NR; ++n)
        rbf[e * NR + n] = scale * sinf((float)(n + 1) * PI_F * ds);
}

// ---------------------------------------------------------------------------
// Kernel 2: out = rbf @ W_rbf + b_rbf   (float4-vectorized, E*32 threads)
// ---------------------------------------------------------------------------
__global__ void out_init_kernel(const float* __restrict__ rbf,
                                const float* __restrict__ W_rbf,
                                const float* __restrict__ b_rbf,
                                float* __restrict__ out, int E) {
    int idx = blockIdx.x * blockDim.x + threadIdx.x;
    if (idx >= E * (H / 4)) return;
    int e  = idx >> 5;          // H/4 == 32 column-groups per edge
    int c4 = (idx & 31) << 2;
    float r[NR];
#pragma unroll
    for (int n = 0; n < NR; ++n) r[n] = rbf[e * NR + n];
    float4 acc = *(const float4*)(b_rbf + c4);
#pragma unroll
    for (int n = 0; n < NR; ++n) {
        float4 w = *(const float4*)(W_rbf + n * H + c4);
        acc.x = fmaf(r[n], w.x, acc.x);
        acc.y = fmaf(r[n], w.y, acc.y);
        acc.z = fmaf(r[n], w.z, acc.z);
        acc.w = fmaf(r[n], w.w, acc.w);
    }
    *(float4*)(out + e * H + c4) = acc;
}

// ---------------------------------------------------------------------------
// Kernel 3: triplet SBF -> fp32 WMMA (16x16x4) -> atomic scatter-add into out.
// One workgroup = 128 threads (4 waves) handles 64 triplets x 128 columns.
// ---------------------------------------------------------------------------
__global__ __launch_bounds__(128) void triplet_wmma_kernel(
    const float* __restrict__ node_pos, const float* __restrict__ group_pos,
    const float* __restrict__ W_sbf, const float* __restrict__ rbf,
    const int* __restrict__ ti, const int* __restrict__ tj,
    const int* __restrict__ tk, const int* __restrict__ id_exp,
    const int* __restrict__ id_red, float* __restrict__ out, int T) {

    __shared__ float Wlds[KPAD][H];     // B matrix, K-padded (rows 42,43 = 0)
    __shared__ float Albs[64][KPAD];    // 64 sbf rows (A matrix), K-padded
    __shared__ int   eids[64];          // id_reduce_ji per triplet row

    const int tid = threadIdx.x;

    // Stage W_sbf (42x128) into LDS with zero padding to 44 rows.
    for (int i = tid; i < KPAD * H; i += 128) {
        int r = i >> 7, c = i & (H - 1);
        Wlds[r][c] = (r < KDIM) ? W_sbf[r * H + c] : 0.0f;
    }

    const int base = blockIdx.x * 64;

    // Phase 1: threads 0..63 each build one sbf row (angle -> cbf, gather rbf).
    if (tid < 64) {
        int t = base + tid;
        int eid = 0;
        if (t < T) {
            eid = id_red[t];
            int gi = ti[t], nj = tj[t], nk = tk[t];
            float cx = group_pos[3 * gi + 0];
            float cy = group_pos[3 * gi + 1];
            float cz = group_pos[3 * gi + 2];
            float r1x = node_pos[3 * nj + 0] - cx;
            float r1y = node_pos[3 * nj + 1] - cy;
            float r1z = node_pos[3 * nj + 2] - cz;
            float r2x = node_pos[3 * nk + 0] - cx;
            float r2y = node_pos[3 * nk + 1] - cy;
            float r2z = node_pos[3 * nk + 2] - cz;
            float n1 = sqrtf(r1x * r1x + r1y * r1y + r1z * r1z) + EPSF;
            float n2 = sqrtf(r2x * r2x + r2y * r2y + r2z * r2z) + EPSF;
            bool close = (n1 < 1e-4f) || (n2 < 1e-4f);
            float i1 = 1.0f / n1, i2 = 1.0f / n2;
            float u1x = r1x * i1, u1y = r1y * i1, u1z = r1z * i1;
            float u2x = r2x * i2, u2y = r2y * i2, u2z = r2z * i2;
            float dp = u1x * u2x + u1y * u2y + u1z * u2z;
            float x = fminf(fmaxf(dp, -1.0f + EPSF), 1.0f - EPSF);
            float crx = u1y * u2z - u1z * u2y;
            float cry = u1z * u2x - u1x * u2z;
            float crz = u1x * u2y - u1y * u2x;
            float y = sqrtf(crx * crx + cry * cry + crz * crz);
            float ang = atan2f(y, x);
            if (close) ang = 0.5f * PI_F;
            ang = fminf(fmaxf(ang, EPSF), PI_F - EPSF);

            float c  = cosf(ang);
            float c2 = c * c, c3 = c2 * c;
            float P[NS];
            P[0] = 1.0f;
            P[1] = c;
            P[2] = (3.0f * c2 - 1.0f) * 0.5f;
            P[3] = (5.0f * c3 - 3.0f * c) * 0.5f;
            P[4] = (35.0f * c2 * c2 - 30.0f * c2 + 3.0f) * 0.125f;
            P[5] = (63.0f * c3 * c2 - 70.0f * c3 + 15.0f * c) * 0.125f;
            P[6] = cosf(6.0f * ang);
            const float coef[NS] = {0.28209479177f, 0.48860251190f,
                                    0.63078313051f, 0.74635266518f,
                                    0.84628437532f, 0.93560257962f,
                                    1.01710723628f};
            int ek = id_exp[t];
            float re[NR];
#pragma unroll
            for (int n = 0; n < NR; ++n) re[n] = rbf[ek * NR + n];
#pragma unroll
            for (int l = 0; l < NS; ++l) {
                float cl = coef[l] * P[l];
#pragma unroll
                for (int n = 0; n < NR; ++n)
                    Albs[tid][l * NR + n] = cl * re[n];
            }
            Albs[tid][42] = 0.0f;
            Albs[tid][43] = 0.0f;
        } else {
#pragma unroll
            for (int k = 0; k < KPAD; ++k) Albs[tid][k] = 0.0f;
        }
        eids[tid] = eid;
    }
    __syncthreads();

    // Phase 2: WMMA. 4 row-blocks x 8 N-tiles = 32 (blk,nt) pairs over 4 waves.
    const int wave = tid >> 5;
    const int lane = tid & 31;
    const int half = lane >> 4;   // 0: K=0,1 ; 1: K=2,3 of each 4-step
    const int l15  = lane & 15;

    for (int p = wave; p < 32; p += 4) {
        int blk = p >> 3;   // triplet row block (16 rows)
        int nt  = p & 7;    // output column tile (16 cols)
        v8f acc = {};
#pragma unroll
        for (int kk = 0; kk < KPAD; kk += 4) {
            v2f a, b;
            // A 16x4 fp32 fragment: v0 = K=(kk+2*half), v1 = K=(kk+2*half+1)
            a.x = Albs[blk * 16 + l15][kk + 2 * half];
            a.y = Albs[blk * 16 + l15][kk + 2 * half + 1];
            // B 4x16 fp32 fragment, K-striped across lane halves
            b.x = Wlds[kk + 2 * half][nt * 16 + l15];
            b.y = Wlds[kk + 2 * half + 1][nt * 16 + l15];
            acc = __builtin_amdgcn_wmma_f32_16x16x4_f32(
                false, a, false, b, (short)0, acc, false, false);
        }
        // C/D layout: lane -> N = l15, VGPR r -> M = r + 8*half
        int col = nt * 16 + l15;
#pragma unroll
        for (int r = 0; r < 8; ++r) {
            int m = blk * 16 + r + 8 * half;
            atomicAdd(&out[eids[m] * H + col], acc[r]);
        }
    }
}

// ---------------------------------------------------------------------------
extern "C" void kernel_launch(void* const* d_in, const int* in_sizes, int n_in,
                              void* d_out, int out_size, void* d_ws, size_t ws_size,
                              hipStream_t stream) {
    const float* node_pos  = (const float*)d_in[0];
    const float* group_pos = (const float*)d_in[1];
    // d_in[2] = edge_attr : unused by the reference
    const float* W_rbf = (const float*)d_in[3];
    const float* b_rbf = (const float*)d_in[4];
    const float* W_sbf = (const float*)d_in[5];
    const int* edge_i  = (const int*)d_in[6];
    const int* edge_j  = (const int*)d_in[7];
    const int* tri_i   = (const int*)d_in[8];
    const int* tri_j   = (const int*)d_in[9];
    const int* tri_k   = (const int*)d_in[10];
    const int* id_exp  = (const int*)d_in[11];
    const int* id_red  = (const int*)d_in[12];

    const int E = in_sizes[6];
    const int T = in_sizes[8];

    float* out = (float*)d_out;
    float* rbf = (float*)d_ws;  // E * NR floats of scratch

    rbf_kernel<<<(E + 255) / 256, 256, 0, stream>>>(node_pos, edge_i, edge_j, rbf, E);

    int initThreads = E * (H / 4);
    out_init_kernel<<<(initThreads + 255) / 256, 256, 0, stream>>>(rbf, W_rbf, b_rbf, out, E);

    int nwg = (T + 63) / 64;
    triplet_wmma_kernel<<<nwg, 128, 0, stream>>>(node_pos, group_pos, W_sbf, rbf,
                                                 tri_i, tri_j, tri_k,
                                                 id_exp, id_red, out, T);
}